// FraudGNN_14869176778811
// MI455X (gfx1250) — compile-verified
//
#include <hip/hip_runtime.h>
#include <hip/hip_bf16.h>

#define N_NODES   100000
#define D         128
#define N_CLASSES 2
#define N_GRAPHS  64

typedef __attribute__((ext_vector_type(2))) float v2f;
typedef __attribute__((ext_vector_type(8))) float v8f;

// ---------------------------------------------------------------------------
// Utility fills
// ---------------------------------------------------------------------------
__global__ void k_fill(float* __restrict__ p, float v, int n) {
    int i = blockIdx.x * blockDim.x + threadIdx.x;
    if (i < n) p[i] = v;
}

__global__ void k_bias_init(float* __restrict__ agg, const float* __restrict__ b) {
    int gid = blockIdx.x * blockDim.x + threadIdx.x;
    if (gid < N_NODES * D) agg[gid] = b[gid & (D - 1)];
}

// ---------------------------------------------------------------------------
// Degree: deg pre-filled with 1.0 (self loop); add 1 per incoming edge.
// Then deg -> rsqrt(deg)  (deg >= 1 always, matches reference dinv).
// ---------------------------------------------------------------------------
__global__ void k_degree(const int* __restrict__ dst, float* __restrict__ deg, int ne) {
    int e = blockIdx.x * blockDim.x + threadIdx.x;
    if (e < ne) atomicAdd(&deg[dst[e]], 1.0f);
}

__global__ void k_dinv(float* __restrict__ deg, int n) {
    int i = blockIdx.x * blockDim.x + threadIdx.x;
    if (i < n) deg[i] = rsqrtf(deg[i]);
}

// ---------------------------------------------------------------------------
// Y[nrows x 128] = X[nrows x 128] @ W[128 x 128], fp32 WMMA 16x16x4.
// One wave per 16x16 C tile; block = 8 waves = one 16-row strip x all columns.
// A layout (16x4 f32): lanes 0-15 hold M=lane, K={k,k+1}; lanes 16-31 K={k+2,k+3}.
// B layout (4x16 f32): lane&15 = N column; half-wave selects K pair in VGPRs.
// C layout: VGPR r -> M = r + 8*(lane>>4), N = lane&15.
// ---------------------------------------------------------------------------
__global__ void __launch_bounds__(256)
k_gemm128_wmma(const float* __restrict__ X, const float* __restrict__ W,
               float* __restrict__ Y, int nrows /* multiple of 16 */) {
    const int lane = threadIdx.x & 31;
    const int wave = threadIdx.x >> 5;      // 0..7 -> column tile
    const int row0 = blockIdx.x * 16;
    const int col0 = wave * 16;
    const int m    = lane & 15;
    const int half = lane >> 4;             // 0 or 1

    const float* arow = X + (size_t)(row0 + m) * D + half * 2;
    const float* bptr = W + (size_t)(half * 2) * D + col0 + m;

    v8f acc = {};
    #pragma unroll 4
    for (int k = 0; k < D; k += 4) {
        v2f a = *(const v2f*)(arow + k);                   // K = k+2*half, k+2*half+1
        v2f b;
        b.x = bptr[(size_t)k * D];                          // W[k+2*half    ][col]
        b.y = bptr[(size_t)k * D + D];                      // W[k+2*half + 1][col]
        acc = __builtin_amdgcn_wmma_f32_16x16x4_f32(
                  /*neg_a=*/false, a, /*neg_b=*/false, b,
                  /*c_mod=*/(short)0, acc, /*reuse_a=*/false, /*reuse_b=*/false);
    }

    float* yrow = Y + (size_t)(row0 + half * 8) * D + col0 + m;
    #pragma unroll
    for (int r = 0; r < 8; ++r)
        yrow[(size_t)r * D] = acc[r];
}

// ---------------------------------------------------------------------------
// Edge scatter: one wave per edge, lane handles 4 features (float4 gather
// from L2-resident h, 4 global_atomic_add_f32 into L2-resident agg).
// norm = dinv[src] * dinv[dst], computed on the fly (wave-uniform loads).
// ---------------------------------------------------------------------------
__global__ void __launch_bounds__(256)
k_edge_scatter(const int* __restrict__ src, const int* __restrict__ dst,
               const float* __restrict__ dinv, const float* __restrict__ h,
               float* __restrict__ agg, int ne) {
    long gid  = (long)blockIdx.x * blockDim.x + threadIdx.x;
    int  e    = (int)(gid >> 5);
    int  f    = ((int)gid & 31) << 2;        // feature offset 0..124
    if (e >= ne) return;
    int s = src[e], d = dst[e];
    float nrm = dinv[s] * dinv[d];
    float4 hv = *(const float4*)(h + (size_t)s * D + f);
    float* out = agg + (size_t)d * D + f;
    atomicAdd(out + 0, hv.x * nrm);
    atomicAdd(out + 1, hv.y * nrm);
    atomicAdd(out + 2, hv.z * nrm);
    atomicAdd(out + 3, hv.w * nrm);
}

// ---------------------------------------------------------------------------
// Self-loop term + optional ReLU, contention-free per-node pass:
// agg[i,f] += dinv[i]^2 * h[i,f];  agg = max(agg, 0) if relu.
// ---------------------------------------------------------------------------
__global__ void k_finalize(const float* __restrict__ h, const float* __restrict__ dinv,
                           float* __restrict__ agg, int relu) {
    int gid = blockIdx.x * blockDim.x + threadIdx.x;
    if (gid >= N_NODES * D) return;
    float di = dinv[gid >> 7];
    float v  = agg[gid] + di * di * h[gid];
    if (relu) v = fmaxf(v, 0.0f);
    agg[gid] = v;
}

// ---------------------------------------------------------------------------
// Mean-pool accumulation (pooled pre-zeroed) + per-graph counts.
// ---------------------------------------------------------------------------
__global__ void k_pool(const float* __restrict__ hfin, const int* __restrict__ batch,
                       float* __restrict__ pooled, float* __restrict__ cnt) {
    int gid = blockIdx.x * blockDim.x + threadIdx.x;
    if (gid >= N_NODES * D) return;
    int i = gid >> 7, f = gid & (D - 1);
    int g = batch[i];
    atomicAdd(&pooled[g * D + f], hfin[gid]);
    if (f == 0) atomicAdd(&cnt[g], 1.0f);
}

// ---------------------------------------------------------------------------
// out[g,c] = (pooled[g]/max(cnt,1)) @ Wc + bc  — 128 threads total.
// ---------------------------------------------------------------------------
__global__ void k_classifier(const float* __restrict__ pooled, const float* __restrict__ cnt,
                             const float* __restrict__ Wc, const float* __restrict__ bc,
                             float* __restrict__ out) {
    int t = threadIdx.x;
    if (t >= N_GRAPHS * N_CLASSES) return;
    int g = t >> 1, c = t & 1;
    float s = 0.0f;
    #pragma unroll 8
    for (int k = 0; k < D; ++k)
        s += pooled[g * D + k] * Wc[k * N_CLASSES + c];
    out[g * N_CLASSES + c] = s / fmaxf(cnt[g], 1.0f) + bc[c];
}

// ---------------------------------------------------------------------------
extern "C" void kernel_launch(void* const* d_in, const int* in_sizes, int n_in,
                              void* d_out, int out_size, void* d_ws, size_t ws_size,
                              hipStream_t stream) {
    const float* x     = (const float*)d_in[0];
    const int*   edges = (const int*)  d_in[1];
    const int*   batch = (const int*)  d_in[2];
    const float* W1    = (const float*)d_in[3];
    const float* b1    = (const float*)d_in[4];
    const float* W2    = (const float*)d_in[5];
    const float* b2    = (const float*)d_in[6];
    const float* Wc    = (const float*)d_in[7];
    const float* bc    = (const float*)d_in[8];
    float*       out   = (float*)d_out;

    const int E = in_sizes[1] / 2;          // edge_index is [2, E] flat
    const int* src = edges;
    const int* dst = edges + E;

    // Workspace carve-up (floats): h | agg | deg/dinv | pooled | cnt
    float* h      = (float*)d_ws;
    float* agg    = h      + (size_t)N_NODES * D;
    float* deg    = agg    + (size_t)N_NODES * D;
    float* pooled = deg    + N_NODES;
    float* cnt    = pooled + N_GRAPHS * D;

    const int T = 256;
    const int ND = N_NODES * D;
    auto blk = [](long n, int t) { return (int)((n + t - 1) / t); };

    // ---- normalization: deg (incl. self loop) -> dinv -----------------------
    k_fill  <<<blk(N_NODES, T), T, 0, stream>>>(deg, 1.0f, N_NODES);
    k_degree<<<blk(E, T),       T, 0, stream>>>(dst, deg, E);
    k_dinv  <<<blk(N_NODES, T), T, 0, stream>>>(deg, N_NODES);   // deg now dinv

    // ---- layer 1: h = x @ W1; agg = scatter + b1; relu ----------------------
    k_gemm128_wmma<<<N_NODES / 16, T, 0, stream>>>(x, W1, h, N_NODES);
    k_bias_init   <<<blk(ND, T), T, 0, stream>>>(agg, b1);
    k_edge_scatter<<<blk((long)E * 32, T), T, 0, stream>>>(src, dst, deg, h, agg, E);
    k_finalize    <<<blk(ND, T), T, 0, stream>>>(h, deg, agg, /*relu=*/1);

    // ---- layer 2: h = agg @ W2; agg = scatter + b2 --------------------------
    k_gemm128_wmma<<<N_NODES / 16, T, 0, stream>>>(agg, W2, h, N_NODES);
    k_bias_init   <<<blk(ND, T), T, 0, stream>>>(agg, b2);
    k_edge_scatter<<<blk((long)E * 32, T), T, 0, stream>>>(src, dst, deg, h, agg, E);
    k_finalize    <<<blk(ND, T), T, 0, stream>>>(h, deg, agg, /*relu=*/0);

    // ---- mean pool + classifier --------------------------------------------
    k_fill<<<blk(N_GRAPHS * D + N_GRAPHS, T), T, 0, stream>>>(pooled, 0.0f,
                                                              N_GRAPHS * D + N_GRAPHS);
    k_pool<<<blk(ND, T), T, 0, stream>>>(agg, batch, pooled, cnt);
    k_classifier<<<1, 128, 0, stream>>>(pooled, cnt, Wc, bc, out);
}